// TprRnn_24696061952431
// MI455X (gfx1250) — compile-verified
//
#include <hip/hip_runtime.h>
#include <math.h>

typedef __attribute__((ext_vector_type(2))) float v2f;
typedef __attribute__((ext_vector_type(8))) float v8f;

#define SYMD 128
#define HIDD 256
#define ED 64
#define RD 32
#define BD 64
#define SD 64
#define WD 12
#define VD 32000

// ---------------------------------------------------------------------------
// 1) Embedding gather + position-weighted sum.  One block per sentence/query
//    row, 128 threads = one per SYM channel (coalesced 512B rows).
// ---------------------------------------------------------------------------
__global__ void embed_kernel(const int* __restrict__ story,
                             const int* __restrict__ query,
                             const float* __restrict__ wemb,
                             const float* __restrict__ pemb,
                             float* __restrict__ sent,
                             float* __restrict__ qsum) {
  int idx = blockIdx.x;
  int t = threadIdx.x;  // 0..127
  float acc = 0.f;
  if (idx < BD * SD) {
    const int* toks = story + (size_t)idx * WD;
    for (int w = 0; w < WD; ++w) {
      int tok = toks[w];
      acc += wemb[(size_t)tok * SYMD + t] * pemb[w * SYMD + t];
    }
    sent[(size_t)idx * SYMD + t] = acc;
  } else {
    int b = idx - BD * SD;
    const int* toks = query + (size_t)b * WD;
    for (int w = 0; w < WD; ++w) {
      int tok = toks[w];
      acc += wemb[(size_t)tok * SYMD + t] * pemb[w * SYMD + t];
    }
    qsum[(size_t)b * SYMD + t] = acc;
  }
}

// ---------------------------------------------------------------------------
// 2) Fused 2-layer MLP via V_WMMA_F32_16X16X4_F32.
//    Block = 128 thr (4 waves) handles 16 rows of X for one head (grid.y).
//    GEMM1 [16,128]@[128,256] -> tanh in LDS -> GEMM2 [16,256]@[256,D].
// ---------------------------------------------------------------------------
__global__ void mlp_wmma_kernel(const float* __restrict__ X,
                                const float* __restrict__ W1,
                                const float* __restrict__ b1,
                                const float* __restrict__ W2,
                                const float* __restrict__ b2,
                                float* __restrict__ Out,
                                int M, int D) {
  __shared__ float Xs[16 * SYMD];   // 8 KB  A-tile
  __shared__ float Hs[16 * HIDD];   // 16 KB hidden tile
  int head = blockIdx.y;
  int row0 = blockIdx.x * 16;
  const float* W1h = W1 + (size_t)head * SYMD * HIDD;
  const float* b1h = b1 + (size_t)head * HIDD;
  const float* W2h = W2 + (size_t)head * HIDD * D;
  const float* b2h = b2 + (size_t)head * D;
  float* Outh = Out + (size_t)head * M * D;
  int tid = threadIdx.x;  // 0..127
  for (int i = 0; i < 16; ++i) {
    int e = tid + i * 128;
    Xs[e] = X[(size_t)(row0 + (e >> 7)) * SYMD + (e & 127)];
  }
  __syncthreads();
  int wave = tid >> 5, lane = tid & 31;
  int nlo = lane & 15;
  int kb = (lane >> 4) * 2;   // A/B k-pair select
  int mb = (lane >> 4) * 8;   // C/D row-half select
  int m = nlo;                // A-fragment row
  // GEMM1: each wave owns 4 N-tiles (64 columns)
  for (int t = 0; t < 4; ++t) {
    int n = wave * 64 + t * 16 + nlo;
    v8f c = {};
    for (int k = 0; k < SYMD; k += 4) {
      v2f a, bfr;
      a.x = Xs[m * SYMD + k + kb];
      a.y = Xs[m * SYMD + k + kb + 1];
      bfr.x = W1h[(size_t)(k + kb) * HIDD + n];
      bfr.y = W1h[(size_t)(k + kb + 1) * HIDD + n];
      c = __builtin_amdgcn_wmma_f32_16x16x4_f32(false, a, false, bfr,
                                                (short)0, c, false, false);
    }
    float bias = b1h[n];
    for (int i = 0; i < 8; ++i)
      Hs[(mb + i) * HIDD + n] = tanhf(c[i] + bias);
  }
  __syncthreads();
  // GEMM2: D/16 output tiles, one per wave (wave-uniform branch: EXEC all-1)
  int nt2 = D >> 4;
  if (wave < nt2) {
    int n = wave * 16 + nlo;
    v8f c = {};
    for (int k = 0; k < HIDD; k += 4) {
      v2f a, bfr;
      a.x = Hs[m * HIDD + k + kb];
      a.y = Hs[m * HIDD + k + kb + 1];
      bfr.x = W2h[(size_t)(k + kb) * D + n];
      bfr.y = W2h[(size_t)(k + kb + 1) * D + n];
      c = __builtin_amdgcn_wmma_f32_16x16x4_f32(false, a, false, bfr,
                                                (short)0, c, false, false);
    }
    float bias = b2h[n];
    for (int i = 0; i < 8; ++i)
      Outh[(size_t)(row0 + mb + i) * D + n] = c[i] + bias;
  }
}

// ---------------------------------------------------------------------------
// 3) Sequential TPR scan.  Block = (batch b, f-slice of 16).  TPR slice
//    [64 e][32 r][16 f] = 128 KB lives in (dynamic) LDS for all 64 steps —
//    CDNA5's 320 KB WGP LDS makes the whole recurrence HBM-free.
//    Thread -> two (r,f) pairs; contractions over e, reductions over r in LDS.
// ---------------------------------------------------------------------------
__global__ void scan_kernel(const float* __restrict__ eo,
                            const float* __restrict__ ro,
                            float* __restrict__ TPRg) {
  extern __shared__ float tprS[];     // 64*512 floats = 128 KB
  __shared__ float sv[224];           // e1(64) e2(64) r1(32) r2(32) r3(32)
  __shared__ float t1a[512], t2a[512];
  __shared__ float whS[16], mhS[16], bhS[16];
  int fs = blockIdx.x;   // f-slice 0..3
  int b = blockIdx.y;
  int tid = threadIdx.x; // 0..255
  const float* e1g = eo;
  const float* e2g = eo + (size_t)4096 * ED;
  const float* r1g = ro;
  const float* r2g = ro + (size_t)4096 * RD;
  const float* r3g = ro + (size_t)2 * 4096 * RD;
  int idx0 = tid, idx1 = tid + 256;
  for (int e = 0; e < ED; ++e) {
    tprS[e * 512 + idx0] = 0.f;
    tprS[e * 512 + idx1] = 0.f;
  }
  int r0 = idx0 >> 4, f0 = idx0 & 15;
  int r1i = idx1 >> 4, f1 = idx1 & 15;
  int fg0 = fs * 16 + f0, fg1 = fs * 16 + f1;
  __syncthreads();
  for (int s = 0; s < SD; ++s) {
    size_t row = (size_t)b * SD + s;
    if (tid < 64)        sv[tid] = e1g[row * ED + tid];
    else if (tid < 128)  sv[tid] = e2g[row * ED + (tid - 64)];
    else if (tid < 160)  sv[tid] = r1g[row * RD + (tid - 128)];
    else if (tid < 192)  sv[tid] = r2g[row * RD + (tid - 160)];
    else if (tid < 224)  sv[tid] = r3g[row * RD + (tid - 192)];
    __syncthreads();
    // t1[r,f] = sum_e e1[e]*TPR ; t2[r,f] = sum_e e2[e]*TPR
    float t1_0 = 0.f, t2_0 = 0.f, t1_1 = 0.f, t2_1 = 0.f;
    for (int e = 0; e < ED; ++e) {
      float e1v = sv[e], e2v = sv[64 + e];
      float te0 = tprS[e * 512 + idx0];
      float te1 = tprS[e * 512 + idx1];
      t1_0 += e1v * te0; t2_0 += e2v * te0;
      t1_1 += e1v * te1; t2_1 += e2v * te1;
    }
    t1a[idx0] = t1_0; t2a[idx0] = t2_0;
    t1a[idx1] = t1_1; t2a[idx1] = t2_1;
    __syncthreads();
    // w_hat/m_hat/b_hat[f] reductions over r
    if (tid < 16) {
      float wh = 0.f, mh = 0.f, bh = 0.f;
      for (int r = 0; r < RD; ++r) {
        float t1v = t1a[r * 16 + tid];
        wh += sv[128 + r] * t1v;
        mh += sv[160 + r] * t1v;
        bh += sv[192 + r] * t2a[r * 16 + tid];
      }
      whS[tid] = wh; mhS[tid] = mh; bhS[tid] = bh;
    }
    __syncthreads();
    // delta[e,r,f] = e1[e]*(r1(e2f-wh)+r2(wh-mh)) + e2[e]*r3(e1f-bh)
    float c1_0 = sv[128 + r0] * (sv[64 + fg0] - whS[f0]) +
                 sv[160 + r0] * (whS[f0] - mhS[f0]);
    float c2_0 = sv[192 + r0] * (sv[fg0] - bhS[f0]);
    float c1_1 = sv[128 + r1i] * (sv[64 + fg1] - whS[f1]) +
                 sv[160 + r1i] * (whS[f1] - mhS[f1]);
    float c2_1 = sv[192 + r1i] * (sv[fg1] - bhS[f1]);
    for (int e = 0; e < ED; ++e) {
      float e1v = sv[e], e2v = sv[64 + e];
      tprS[e * 512 + idx0] += e1v * c1_0 + e2v * c2_0;
      tprS[e * 512 + idx1] += e1v * c1_1 + e2v * c2_1;
    }
    __syncthreads();
  }
  for (int e = 0; e < ED; ++e) {
    TPRg[(((size_t)b * ED + e) * RD + r0) * ED + fg0] = tprS[e * 512 + idx0];
    TPRg[(((size_t)b * ED + e) * RD + r1i) * ED + fg1] = tprS[e * 512 + idx1];
  }
}

// ---------------------------------------------------------------------------
// 4) Inference: 3 chained (e x r x TPR) contractions + LayerNorm per batch.
//    Block per b; thread = (r-group, f); TPR reads are 256B-coalesced on f.
// ---------------------------------------------------------------------------
__global__ void infer_kernel(const float* __restrict__ TPRg,
                             const float* __restrict__ qe1,
                             const float* __restrict__ qr,
                             const float* __restrict__ ln_g,
                             const float* __restrict__ ln_b,
                             float* __restrict__ U) {
  __shared__ float tin[64], qrS[32], part[4 * 64], vS[64], iS[64], usum[64];
  __shared__ float muS, varS;
  int b = blockIdx.x;
  int tid = threadIdx.x;  // 0..255
  int f = tid & 63, rg = tid >> 6;
  if (tid < 64) { usum[tid] = 0.f; tin[tid] = qe1[(size_t)b * 64 + tid]; }
  __syncthreads();
  const float* tprb = TPRg + (size_t)b * ED * RD * ED;
  for (int j = 0; j < 3; ++j) {
    if (tid >= 64 && tid < 96) {
      int r = tid - 64;
      qrS[r] = qr[((size_t)j * BD + b) * RD + r];
    }
    __syncthreads();
    float acc = 0.f;
    for (int rr = 0; rr < 8; ++rr) {
      int r = rg * 8 + rr;
      float ts = 0.f;
      for (int e = 0; e < ED; ++e)
        ts += tin[e] * tprb[((size_t)e * RD + r) * ED + f];
      acc += qrS[r] * ts;
    }
    part[rg * 64 + f] = acc;
    __syncthreads();
    if (tid < 64)
      vS[tid] = part[tid] + part[64 + tid] + part[128 + tid] + part[192 + tid];
    __syncthreads();
    if (tid == 0) {
      float mu = 0.f;
      for (int i = 0; i < 64; ++i) mu += vS[i];
      mu *= (1.f / 64.f);
      float var = 0.f;
      for (int i = 0; i < 64; ++i) { float d = vS[i] - mu; var += d * d; }
      muS = mu; varS = var * (1.f / 64.f);
    }
    __syncthreads();
    if (tid < 64) {
      float x = (vS[tid] - muS) * rsqrtf(varS + 1e-5f) * ln_g[j * 64 + tid] +
                ln_b[j * 64 + tid];
      iS[tid] = x;
      usum[tid] += x;
    }
    __syncthreads();
    if (tid < 64) tin[tid] = iS[tid];
    __syncthreads();
  }
  if (tid < 64) U[(size_t)b * 64 + tid] = usum[tid];
}

// ---------------------------------------------------------------------------
// 5) Vocab projection [64,64]@[64,32000] via WMMA f32 16x16x4.
//    Block = 128 thr (4 waves) -> 4 consecutive N-tiles of one M-tile.
// ---------------------------------------------------------------------------
__global__ void zproj_kernel(const float* __restrict__ U,
                             const float* __restrict__ Z,
                             float* __restrict__ Out) {
  __shared__ float Us[16 * 64];
  int row0 = blockIdx.y * 16;
  int ng = blockIdx.x;
  int tid = threadIdx.x;  // 0..127
  for (int i = 0; i < 8; ++i) {
    int e = tid + i * 128;
    Us[e] = U[(size_t)(row0 + (e >> 6)) * 64 + (e & 63)];
  }
  __syncthreads();
  int wave = tid >> 5, lane = tid & 31;
  int nlo = lane & 15;
  int kb = (lane >> 4) * 2, mb = (lane >> 4) * 8;
  int m = nlo;
  int n = ng * 64 + wave * 16 + nlo;
  v8f c = {};
  for (int k = 0; k < 64; k += 4) {
    v2f a, bfr;
    a.x = Us[m * 64 + k + kb];
    a.y = Us[m * 64 + k + kb + 1];
    bfr.x = Z[(size_t)(k + kb) * VD + n];
    bfr.y = Z[(size_t)(k + kb + 1) * VD + n];
    c = __builtin_amdgcn_wmma_f32_16x16x4_f32(false, a, false, bfr,
                                              (short)0, c, false, false);
  }
  for (int i = 0; i < 8; ++i)
    Out[(size_t)(row0 + mb + i) * VD + n] = c[i];
}

// ---------------------------------------------------------------------------
extern "C" void kernel_launch(void* const* d_in, const int* in_sizes, int n_in,
                              void* d_out, int out_size, void* d_ws,
                              size_t ws_size, hipStream_t stream) {
  const int* story = (const int*)d_in[0];
  const int* query = (const int*)d_in[1];
  const float* wemb = (const float*)d_in[2];
  const float* pemb = (const float*)d_in[3];
  const float* ue_W1 = (const float*)d_in[4];
  const float* ue_b1 = (const float*)d_in[5];
  const float* ue_W2 = (const float*)d_in[6];
  const float* ue_b2 = (const float*)d_in[7];
  const float* ur_W1 = (const float*)d_in[8];
  const float* ur_b1 = (const float*)d_in[9];
  const float* ur_W2 = (const float*)d_in[10];
  const float* ur_b2 = (const float*)d_in[11];
  const float* ie_W1 = (const float*)d_in[12];
  const float* ie_b1 = (const float*)d_in[13];
  const float* ie_W2 = (const float*)d_in[14];
  const float* ie_b2 = (const float*)d_in[15];
  const float* ir_W1 = (const float*)d_in[16];
  const float* ir_b1 = (const float*)d_in[17];
  const float* ir_W2 = (const float*)d_in[18];
  const float* ir_b2 = (const float*)d_in[19];
  const float* ln_g = (const float*)d_in[20];
  const float* ln_b = (const float*)d_in[21];
  const float* Z = (const float*)d_in[22];
  float* out = (float*)d_out;
  float* ws = (float*)d_ws;
  // workspace layout (floats)
  float* sent = ws;                        // 64*64*128   = 524288
  float* qsum = sent + 524288;             // 64*128      = 8192
  float* eo   = qsum + 8192;               // 2*4096*64   = 524288
  float* ro   = eo + 524288;               // 3*4096*32   = 393216
  float* TPRg = ro + 393216;               // 64*64*32*64 = 8388608
  float* qe1  = TPRg + 8388608;            // 64*64
  float* qr   = qe1 + 4096;                // 3*64*32
  float* U    = qr + 6144;                 // 64*64

  embed_kernel<<<dim3(BD * SD + BD), dim3(128), 0, stream>>>(
      story, query, wemb, pemb, sent, qsum);
  mlp_wmma_kernel<<<dim3(256, 2), dim3(128), 0, stream>>>(
      sent, ue_W1, ue_b1, ue_W2, ue_b2, eo, 4096, 64);
  mlp_wmma_kernel<<<dim3(256, 3), dim3(128), 0, stream>>>(
      sent, ur_W1, ur_b1, ur_W2, ur_b2, ro, 4096, 32);
  scan_kernel<<<dim3(4, 64), dim3(256), 64 * 512 * sizeof(float), stream>>>(
      eo, ro, TPRg);
  mlp_wmma_kernel<<<dim3(4, 1), dim3(128), 0, stream>>>(
      qsum, ie_W1, ie_b1, ie_W2, ie_b2, qe1, 64, 64);
  mlp_wmma_kernel<<<dim3(4, 3), dim3(128), 0, stream>>>(
      qsum, ir_W1, ir_b1, ir_W2, ir_b2, qr, 64, 32);
  infer_kernel<<<dim3(64), dim3(256), 0, stream>>>(
      TPRg, qe1, qr, ln_g, ln_b, U);
  zproj_kernel<<<dim3(500, 4), dim3(128), 0, stream>>>(U, Z, out);
}